// myGCN_3384434230048
// MI455X (gfx1250) — compile-verified
//
#include <hip/hip_runtime.h>

// ---------------------------------------------------------------------------
// 2-layer GCN on MI455X (gfx1250, wave32).
// Memory-roofline bound (~1 GB of gather/scatter traffic -> ~45us floor at
// 23.3 TB/s HBM). Node accumulators (deg/agg1/agg2/h, ~32MB) and the edge
// lists (128MB, reused by both edge passes) all fit in the 192MB L2, so we
// use plain non-returning global_atomic_add_f32 (relaxed, agent scope) for
// the scatter and default RT cache hints for the streams.
// The tiny per-node affine transforms are batched 16 nodes/wave through
// V_WMMA_F32_16X16X4_F32 (A = W^T zero-padded 16x4, B = 4x16 node features).
// ---------------------------------------------------------------------------

typedef float v2f __attribute__((ext_vector_type(2)));
typedef float v8f __attribute__((ext_vector_type(8)));

#define N_NODES 1000000
#define N_EDGES 16000000

static __device__ __forceinline__ void atomic_add_f32(float* p, float v) {
  // Non-returning fp32 atomic add -> GLOBAL_ATOMIC_ADD_F32 (STOREcnt path).
  __hip_atomic_fetch_add(p, v, __ATOMIC_RELAXED, __HIP_MEMORY_SCOPE_AGENT);
}

// -------------------------- workspace zeroing ------------------------------
__global__ void gcn_zero_kernel(float4* __restrict__ p, int n4) {
  int i = blockIdx.x * blockDim.x + threadIdx.x;
  int stride = gridDim.x * blockDim.x;
  float4 z = make_float4(0.f, 0.f, 0.f, 0.f);
  for (; i < n4; i += stride) p[i] = z;
}

// ------------------ edge pass 1: degree + feat scatter ---------------------
__global__ void gcn_edge1_kernel(const int* __restrict__ esrc,
                                 const int* __restrict__ edst,
                                 const float* __restrict__ feat,
                                 float* __restrict__ deg,
                                 float* __restrict__ agg1) {
  int e = blockIdx.x * blockDim.x + threadIdx.x;
  if (e >= N_EDGES) return;
  int s = esrc[e];
  int d = edst[e];
  float f0 = feat[3 * s + 0];
  float f1 = feat[3 * s + 1];
  float f2 = feat[3 * s + 2];
  atomic_add_f32(&deg[d], 1.0f);
  atomic_add_f32(&agg1[3 * d + 0], f0);
  atomic_add_f32(&agg1[3 * d + 1], f1);
  atomic_add_f32(&agg1[3 * d + 2], f2);
}

// ---- node pass 1: h = relu((agg1 * deg_inv) @ W1 + b1), 16 nodes / WMMA ---
// D(16x16) = A(16x4) x B(4x16):
//   A[m][k] = W1[k][m]  (m<2 output dims, k<3 input dims, rest zero)
//   B[k][n] = agg1[node_n][k] * deg_inv[node_n]  (k<3, zero-padded K=3)
// A layout: lane (16*khi + m): {A[m][2*khi], A[m][2*khi+1]}
// B layout: lane (16*khi + n): {B[2*khi][n], B[2*khi+1][n]}
// D layout: VGPR r, lanes 0-15 hold D[r][lane] -> node n's outputs are
//           c[0..1] on lane n (rows >=2 are zero padding).
__global__ void gcn_node1_wmma_kernel(const float* __restrict__ agg1,
                                      const float* __restrict__ deg,
                                      const float* __restrict__ W1,
                                      const float* __restrict__ b1,
                                      float* __restrict__ h,
                                      int ntiles) {
  int wave = (blockIdx.x * blockDim.x + threadIdx.x) >> 5;  // wave-uniform
  if (wave >= ntiles) return;                               // uniform exit
  int lane = threadIdx.x & 31;
  int n = lane & 15;
  int hi = lane >> 4;  // K-half selector
  int node = wave * 16 + n;

  // A = W1^T zero-padded (W1 row-major [3][2]: W1[k][m] = W1[k*2+m])
  v2f a;
  if (hi == 0) {
    a.x = (n < 2) ? W1[0 * 2 + n] : 0.0f;  // K=0
    a.y = (n < 2) ? W1[1 * 2 + n] : 0.0f;  // K=1
  } else {
    a.x = (n < 2) ? W1[2 * 2 + n] : 0.0f;  // K=2
    a.y = 0.0f;                            // K=3 pad
  }

  float dg = deg[node];
  float dv = (dg > 0.0f) ? (1.0f / dg) : 0.0f;

  v2f b;
  if (hi == 0) {
    b.x = agg1[3 * node + 0] * dv;
    b.y = agg1[3 * node + 1] * dv;
  } else {
    b.x = agg1[3 * node + 2] * dv;
    b.y = 0.0f;
  }

  v8f c = {};
  c = __builtin_amdgcn_wmma_f32_16x16x4_f32(
      /*neg_a=*/false, a, /*neg_b=*/false, b,
      /*c_mod=*/(short)0, c, /*reuse_a=*/false, /*reuse_b=*/false);

  if (hi == 0) {
    float h0 = c[0] + b1[0];
    float h1 = c[1] + b1[1];
    h0 = h0 > 0.0f ? h0 : 0.0f;
    h1 = h1 > 0.0f ? h1 : 0.0f;
    ((float2*)h)[node] = make_float2(h0, h1);  // coalesced b64 store
  }
}

// -------------------- edge pass 2: h scatter -------------------------------
__global__ void gcn_edge2_kernel(const int* __restrict__ esrc,
                                 const int* __restrict__ edst,
                                 const float* __restrict__ h,
                                 float* __restrict__ agg2) {
  int e = blockIdx.x * blockDim.x + threadIdx.x;
  if (e >= N_EDGES) return;
  int s = esrc[e];
  int d = edst[e];
  float2 hv = ((const float2*)h)[s];  // aligned b64 gather
  atomic_add_f32(&agg2[2 * d + 0], hv.x);
  atomic_add_f32(&agg2[2 * d + 1], hv.y);
}

// ---- node pass 2: out = (agg2 * deg_inv) @ W2 + b2, 16 nodes / WMMA -------
//   A[m][k] = W2[k][m]  (m<3 output dims, k<2 input dims, rest zero)
//   B[k][n] = agg2[node_n][k] * deg_inv[node_n]  (k<2)
__global__ void gcn_node2_wmma_kernel(const float* __restrict__ agg2,
                                      const float* __restrict__ deg,
                                      const float* __restrict__ W2,
                                      const float* __restrict__ b2,
                                      float* __restrict__ out,
                                      int ntiles) {
  int wave = (blockIdx.x * blockDim.x + threadIdx.x) >> 5;
  if (wave >= ntiles) return;  // wave-uniform exit
  int lane = threadIdx.x & 31;
  int n = lane & 15;
  int hi = lane >> 4;
  int node = wave * 16 + n;

  // A = W2^T zero-padded (W2 row-major [2][3]: W2[k][m] = W2[k*3+m])
  v2f a;
  if (hi == 0) {
    a.x = (n < 3) ? W2[0 * 3 + n] : 0.0f;  // K=0
    a.y = (n < 3) ? W2[1 * 3 + n] : 0.0f;  // K=1
  } else {
    a.x = 0.0f;  // K=2 pad
    a.y = 0.0f;  // K=3 pad
  }

  float dg = deg[node];
  float dv = (dg > 0.0f) ? (1.0f / dg) : 0.0f;

  v2f b;
  if (hi == 0) {
    b.x = agg2[2 * node + 0] * dv;
    b.y = agg2[2 * node + 1] * dv;
  } else {
    b.x = 0.0f;
    b.y = 0.0f;
  }

  v8f c = {};
  c = __builtin_amdgcn_wmma_f32_16x16x4_f32(
      false, a, false, b, (short)0, c, false, false);

  if (hi == 0) {
    out[3 * node + 0] = c[0] + b2[0];
    out[3 * node + 1] = c[1] + b2[1];
    out[3 * node + 2] = c[2] + b2[2];
  }
}

// ---------------------------------------------------------------------------
extern "C" void kernel_launch(void* const* d_in, const int* in_sizes, int n_in,
                              void* d_out, int out_size, void* d_ws, size_t ws_size,
                              hipStream_t stream) {
  const float* feat = (const float*)d_in[0];  // [N,3]
  const float* W1   = (const float*)d_in[1];  // [3,2]
  const float* b1   = (const float*)d_in[2];  // [2]
  const float* W2   = (const float*)d_in[3];  // [2,3]
  const float* b2   = (const float*)d_in[4];  // [3]
  const int* esrc   = (const int*)d_in[5];    // [E]
  const int* edst   = (const int*)d_in[6];    // [E]
  float* out = (float*)d_out;                 // [N,3]

  float* ws   = (float*)d_ws;
  float* deg  = ws;                 // N      floats
  float* agg1 = ws + N_NODES;       // 3N     floats
  float* h    = ws + 4 * N_NODES;   // 2N     floats (8B-aligned offset)
  float* agg2 = ws + 6 * N_NODES;   // 2N     floats (8B-aligned offset)
  // total: 8N floats = 32 MB of workspace

  // Zero the whole accumulator region every call (atomics accumulate).
  int n4 = (8 * N_NODES) / 4;
  gcn_zero_kernel<<<4096, 256, 0, stream>>>((float4*)ws, n4);

  // Edge pass 1: degree + layer-1 message scatter. 16M / 256 = 62500 blocks.
  gcn_edge1_kernel<<<N_EDGES / 256, 256, 0, stream>>>(esrc, edst, feat, deg, agg1);

  // Node pass 1: 62500 tiles of 16 nodes, 8 waves/block -> 7813 blocks.
  int ntiles = N_NODES / 16;  // 62500
  int nblocks = (ntiles + 7) / 8;
  gcn_node1_wmma_kernel<<<nblocks, 256, 0, stream>>>(agg1, deg, W1, b1, h, ntiles);

  // Edge pass 2: layer-2 message scatter.
  gcn_edge2_kernel<<<N_EDGES / 256, 256, 0, stream>>>(esrc, edst, h, agg2);

  // Node pass 2: final transform to d_out.
  gcn_node2_wmma_kernel<<<nblocks, 256, 0, stream>>>(agg2, deg, W2, b2, out, ntiles);
}